// HyperbolicAggregation_74363063763643
// MI455X (gfx1250) — compile-verified
//
#include <hip/hip_runtime.h>
#include <math.h>

typedef __attribute__((ext_vector_type(2))) float v2f;
typedef __attribute__((ext_vector_type(4))) float v4f;
typedef __attribute__((ext_vector_type(8))) float v8f;

#define N_NODES 1024
#define DIM     128
#define EPSF    1e-7f

// ---------------------------------------------------------------------------
// Kernel 1: zero the dense adjacency workspace
// ---------------------------------------------------------------------------
__global__ void zero_kernel(float* __restrict__ p, int n) {
    int i = blockIdx.x * blockDim.x + threadIdx.x;
    if (i < n) p[i] = 0.0f;
}

// ---------------------------------------------------------------------------
// Kernel 2: scatter edges -> dense adjacency (duplicates accumulate, matching
// jnp .at[r,c].add(1.0))
// ---------------------------------------------------------------------------
__global__ void scatter_kernel(const int* __restrict__ adj, float* __restrict__ adjw, int E) {
    int e = blockIdx.x * blockDim.x + threadIdx.x;
    if (e < E) {
        int rr = adj[e];      // adj[0][e]
        int cc = adj[E + e];  // adj[1][e]
        atomicAdd(adjw + (size_t)rr * N_NODES + cc, 1.0f);
    }
}

// ---------------------------------------------------------------------------
// Kernel 3: l_i = log_0(x_i)@W[:D], r_i = log_0(x_i)@W[D:]
// log_0(x) = (arccosh(x0)/||spatial||) * (0, spatial)   -> pure wave reduction
// One wave (32 lanes) per node.
// ---------------------------------------------------------------------------
__global__ __launch_bounds__(128) void lr_kernel(const float* __restrict__ x,
                                                 const float* __restrict__ W,
                                                 float* __restrict__ l,
                                                 float* __restrict__ r) {
    int node = blockIdx.x * (blockDim.x >> 5) + (threadIdx.x >> 5);
    int lane = threadIdx.x & 31;
    const float* xr = x + (size_t)node * DIM;
    float ss = 0.0f, sl = 0.0f, sr = 0.0f;
    for (int d = lane; d < DIM; d += 32) {
        float v = xr[d];
        if (d > 0) {
            ss += v * v;
            sl += v * W[d];
            sr += v * W[DIM + d];
        }
    }
    for (int off = 16; off; off >>= 1) {
        ss += __shfl_xor(ss, off, 32);
        sl += __shfl_xor(sl, off, 32);
        sr += __shfl_xor(sr, off, 32);
    }
    if (lane == 0) {
        float alpha = fmaxf(xr[0], 1.0f + EPSF);
        float dist  = acoshf(alpha);
        float un    = sqrtf(fmaxf(ss, EPSF));
        float f     = dist / un;
        l[node] = f * sl;
        r[node] = f * sr;
    }
}

// ---------------------------------------------------------------------------
// Kernel 4: fused attention + tangent aggregation.
// Per 16-row i-tile (one block, 4 waves splitting the 64 j-tiles):
//   alpha tile  = 2*x0_i*x0_j - (X_i . X_j^T)        (WMMA f32 16x16x4, K=128)
//   w_ij        = adj_ij * sigmoid(l_i+r_j+b) * acosh(a)/sqrt(a^2-1)
//   sup tile   += W(16x16) @ X_j(16x128)             (WMMA f32 16x16x4, K=16)
//   swa[m]     += sum_j w*alpha
// support[i,:] = sup[i,:] - swa[i]*x[i,:]
// ---------------------------------------------------------------------------
__global__ __launch_bounds__(128) void attn_agg_kernel(const float* __restrict__ x,
                                                       const float* __restrict__ adjw,
                                                       const float* __restrict__ l,
                                                       const float* __restrict__ r,
                                                       const float* __restrict__ bptr,
                                                       float* __restrict__ support) {
    __shared__ float s_x[16 * DIM];      // i-tile of X (shared by all 4 waves)
    __shared__ float s_wt[4][16 * 16];   // per-wave weight-tile staging (C->A relayout)
    __shared__ float s_sup[16 * DIM];    // cross-wave support accumulator
    __shared__ float s_swa[16];          // cross-wave sum(w*alpha) per row

    const int tid  = threadIdx.x;
    const int wave = tid >> 5;
    const int lane = tid & 31;
    const int half = lane >> 4;   // 0: rows/K-lo group, 1: hi group
    const int nlan = lane & 15;
    const int i0   = blockIdx.x * 16;
    const float bval = bptr[0];

    // cooperative load of the i-tile (16x128 f32 = 8KB) into LDS
    for (int t = tid; t < 16 * DIM / 4; t += 128) {
        *(v4f*)(s_x + t * 4) = *(const v4f*)(x + (size_t)i0 * DIM + t * 4);
    }
    for (int t = tid; t < 16 * DIM; t += 128) s_sup[t] = 0.0f;
    if (tid < 16) s_swa[tid] = 0.0f;
    __syncthreads();

    // per-lane row constants (8 rows each: m = v + 8*half)
    float li[8], x0i[8];
    for (int v = 0; v < 8; ++v) {
        int m = v + 8 * half;
        li[v]  = l[i0 + m];
        x0i[v] = s_x[m * DIM];
    }

    v8f sup[8];
    for (int t = 0; t < 8; ++t) sup[t] = (v8f)0.0f;
    float swa[8];
    for (int v = 0; v < 8; ++v) swa[v] = 0.0f;

    float* wt = s_wt[wave];

    // each wave takes every 4th j-tile
    for (int j0 = wave * 16; j0 < N_NODES; j0 += 64) {
        // ---- Gram tile: g = X_i(16xK) . X_j^T(Kx16), K=128 in chunks of 4 ----
        v8f g = (v8f)0.0f;
        const float* xjrow = x + (size_t)(j0 + nlan) * DIM;  // A and B frags index identically
        for (int k0 = 0; k0 < DIM; k0 += 4) {
            int kk = k0 + 2 * half;
            v2f a, b;
            a.x = s_x[nlan * DIM + kk];
            a.y = s_x[nlan * DIM + kk + 1];
            b.x = xjrow[kk];
            b.y = xjrow[kk + 1];
            g = __builtin_amdgcn_wmma_f32_16x16x4_f32(false, a, false, b,
                                                      (short)0, g, false, false);
        }

        float x0j = xjrow[0];
        float rj  = r[j0 + nlan];

        // ---- elementwise weights (lane owns (m = v+8*half, n = nlan)) ----
        for (int v = 0; v < 8; ++v) {
            int m = v + 8 * half;
            float alpha = fmaxf(2.0f * x0i[v] * x0j - g[v], 1.0f + EPSF);
            float ac    = adjw[(size_t)(i0 + m) * N_NODES + (j0 + nlan)];
            float z     = li[v] + rj + bval;
            float att   = 1.0f / (1.0f + __expf(-z));
            float dist  = acoshf(alpha);
            float un    = sqrtf(fmaxf(alpha * alpha - 1.0f, EPSF));
            float w     = ac * att * (dist / un);
            swa[v] += w * alpha;
            wt[m * 16 + nlan] = w;   // stage in C-layout row-major
        }
        // wave-local: make staged tile visible to our own DS loads
        asm volatile("s_wait_dscnt 0x0" ::: "memory");

        // ---- aggregation tile: sup += W(16x16) @ X_j(16x128), K=16 ----
        for (int k0 = 0; k0 < 16; k0 += 4) {
            int kk = k0 + 2 * half;
            v2f a2;
            a2.x = wt[nlan * 16 + kk];      // A-frag relayout via LDS
            a2.y = wt[nlan * 16 + kk + 1];
            const float* xb = x + (size_t)(j0 + kk) * DIM + nlan;
            for (int t = 0; t < 8; ++t) {
                v2f b2;
                b2.x = xb[t * 16];          // row j0+kk,   col t*16+nlan
                b2.y = xb[t * 16 + DIM];    // row j0+kk+1, col t*16+nlan
                sup[t] = __builtin_amdgcn_wmma_f32_16x16x4_f32(false, a2, false, b2,
                                                               (short)0, sup[t], false, false);
            }
        }
    }

    // ---- combine the 4 waves' partials via LDS float atomics ----
    for (int t = 0; t < 8; ++t) {
        for (int v = 0; v < 8; ++v) {
            int m = v + 8 * half;
            int d = t * 16 + nlan;
            atomicAdd(&s_sup[m * DIM + d], sup[t][v]);  // ds_add_f32
        }
    }
    for (int v = 0; v < 8; ++v) {
        int m = v + 8 * half;
        atomicAdd(&s_swa[m], swa[v]);
    }
    __syncthreads();

    // ---- support = sup - swa*x_i ; write out ----
    for (int t = tid; t < 16 * DIM; t += 128) {
        int m = t / DIM, d = t % DIM;
        support[(size_t)(i0 + m) * DIM + d] = s_sup[t] - s_swa[m] * s_x[m * DIM + d];
    }
}

// ---------------------------------------------------------------------------
// Kernel 5: out = proj(exp_map(x, support)). One wave per node.
// ---------------------------------------------------------------------------
__global__ __launch_bounds__(128) void expmap_kernel(const float* __restrict__ x,
                                                     const float* __restrict__ sup,
                                                     float* __restrict__ out) {
    int node = blockIdx.x * (blockDim.x >> 5) + (threadIdx.x >> 5);
    int lane = threadIdx.x & 31;
    const float* xr = x + (size_t)node * DIM;
    const float* ur = sup + (size_t)node * DIM;
    float xu[4], uu[4], s = 0.0f;
    for (int t = 0; t < 4; ++t) {
        int d = lane + 32 * t;
        xu[t] = xr[d];
        uu[t] = ur[d];
        s += uu[t] * uu[t];
    }
    for (int off = 16; off; off >>= 1) s += __shfl_xor(s, off, 32);
    float u0 = __shfl(uu[0], 0, 32);
    float md = s - 2.0f * u0 * u0;               // Minkowski <u,u>
    float un = sqrtf(fmaxf(md, EPSF));
    float ch = coshf(un);
    float sh = sinhf(un) / un;
    float y[4], s2 = 0.0f;
    for (int t = 0; t < 4; ++t) {
        int d = lane + 32 * t;
        y[t] = ch * xu[t] + sh * uu[t];
        if (d > 0) s2 += y[t] * y[t];
    }
    for (int off = 16; off; off >>= 1) s2 += __shfl_xor(s2, off, 32);
    for (int t = 0; t < 4; ++t) {
        int d = lane + 32 * t;
        out[(size_t)node * DIM + d] = (d == 0) ? sqrtf(1.0f + s2) : y[t];
    }
}

// ---------------------------------------------------------------------------
extern "C" void kernel_launch(void* const* d_in, const int* in_sizes, int n_in,
                              void* d_out, int out_size, void* d_ws, size_t ws_size,
                              hipStream_t stream) {
    const float* x  = (const float*)d_in[0];   // [N, D] f32
    const float* W  = (const float*)d_in[1];   // [2D]   f32
    const float* b  = (const float*)d_in[2];   // [1]    f32
    const int*   aj = (const int*)d_in[3];     // [2, E] i32
    float* out = (float*)d_out;                // [N, D] f32

    // workspace layout (floats): adj dense | l | r | support
    float* adjw = (float*)d_ws;                                  // N*N
    float* l    = adjw + (size_t)N_NODES * N_NODES;              // N
    float* r    = l + N_NODES;                                   // N
    float* sup  = r + N_NODES;                                   // N*D
    const int E = in_sizes[3] / 2;

    zero_kernel<<<(N_NODES * N_NODES + 255) / 256, 256, 0, stream>>>(adjw, N_NODES * N_NODES);
    scatter_kernel<<<(E + 255) / 256, 256, 0, stream>>>(aj, adjw, E);
    lr_kernel<<<N_NODES / 4, 128, 0, stream>>>(x, W, l, r);
    attn_agg_kernel<<<N_NODES / 16, 128, 0, stream>>>(x, adjw, l, r, b, sup);
    expmap_kernel<<<N_NODES / 4, 128, 0, stream>>>(x, sup, out);
}